// DeTokenizer_23716809408981
// MI455X (gfx1250) — compile-verified
//
#include <hip/hip_runtime.h>
#include <hip/hip_bf16.h>
#include <stdint.h>

typedef __attribute__((ext_vector_type(2))) float v2f;
typedef __attribute__((ext_vector_type(8))) float v8f;

#define B_ 8
#define L_ 4096
#define M_ 2048
#define D_ 512
#define S_ 16                 // segments along M
#define SEGM_ (M_ / S_)       // 128 rows per segment
#define SEGCHUNKS_ (SEGM_ / 16)  // 8 wmma chunks per segment

// ---------------------------------------------------------------------------
// Phase A: per-batch compaction. One block per batch row, 256 threads.
//  - inclusive prefix sum of token_mask over L (block scan in LDS)
//  - cidx[l] = cumsum-1 for every position
//  - scatter clip(1-prob) into decay slots; invalid slots get decay=1.0
// ---------------------------------------------------------------------------
__global__ void phaseA_compact(const float* __restrict__ prob,
                               const unsigned char* __restrict__ mask,
                               float* __restrict__ decayBuf,
                               int* __restrict__ cidxBuf) {
  const int b = blockIdx.x;
  const int t = threadIdx.x;  // 256 threads
  __shared__ int sums[256];

  for (int m = t; m < M_; m += 256) decayBuf[b * M_ + m] = 1.0f;

  const int base = b * L_ + t * 16;
  unsigned char mv[16];
  int s = 0;
#pragma unroll
  for (int i = 0; i < 16; ++i) {
    mv[i] = mask[base + i];
    s += mv[i];
  }
  sums[t] = s;
  __syncthreads();
  for (int off = 1; off < 256; off <<= 1) {
    int v = (t >= off) ? sums[t - off] : 0;
    __syncthreads();
    sums[t] += v;
    __syncthreads();
  }
  int run = sums[t] - s;  // exclusive prefix
#pragma unroll
  for (int i = 0; i < 16; ++i) {
    run += (int)mv[i];
    cidxBuf[base + i] = run - 1;
    if (mv[i]) {
      float dec = 1.0f - prob[base + i];
      dec = fminf(fmaxf(dec, 0.0f), 1.0f);
      decayBuf[b * M_ + (run - 1)] = dec;
    }
  }
}

// ---------------------------------------------------------------------------
// Phase A2: per (b, segment) inclusive decay cumprod within the segment, plus
// the full-segment product. Scalar per m (independent of D) -> 128 threads.
// ---------------------------------------------------------------------------
__global__ void phaseA2_cumprod(const float* __restrict__ decayBuf,
                                float* __restrict__ cumseg,
                                float* __restrict__ Pseg) {
  const int idx = threadIdx.x;  // B_*S_ = 128
  const int b = idx / S_;
  const int s = idx % S_;
  const int base = b * M_ + s * SEGM_;
  float p = 1.0f;
  for (int i = 0; i < SEGM_; ++i) {
    p *= decayBuf[base + i];
    cumseg[base + i] = p;
  }
  Pseg[b * S_ + s] = p;
}

// ---------------------------------------------------------------------------
// Phase B1: segmented chunked EMA via f32 WMMA.
// One wave per (batch, 16-col d-tile, segment) -> 4096 independent waves,
// each only SEGCHUNKS_=8 serial chunk steps. Local scan (carry-in = 0);
// stores UNMASKED local h; stores segment-final h for the combine pass.
// Per chunk: h = T(16x16 lower-tri) @ Bmat(16x16) + cumprod * carry,
// T@Bmat = 4x V_WMMA_F32_16X16X4_F32 (K=4 accumulation).
// ---------------------------------------------------------------------------
__global__ void __launch_bounds__(32) phaseB1_ema_wmma(
    const float* __restrict__ hidden, const float* __restrict__ decayBuf,
    float* __restrict__ emaBuf, float* __restrict__ segfinal) {
  const int seg = blockIdx.x & (S_ - 1);
  const int dt = (blockIdx.x >> 4) & 31;   // D_/16 == 32 tiles
  const int b = blockIdx.x >> 9;
  const int d0 = dt * 16;
  const int lane = threadIdx.x;
  const int row = lane & 15;  // matrix row (A) / column n (B,C,D)
  const int hi = lane >> 4;   // upper half-wave selector
  const size_t hBase = (size_t)b * M_ * D_;
  const int segBase = seg * SEGM_;

  float carry = 0.0f;  // local scan: global carry applied in fixup pass

  for (int cb = 0; cb < SEGCHUNKS_; ++cb) {
    const int m0 = segBase + cb * 16;

    // 16 decays for this chunk (uniform addresses -> one s_load_b512)
    float a[16];
#pragma unroll
    for (int k = 0; k < 16; ++k) a[k] = decayBuf[b * M_ + m0 + k];

    // inclusive cumulative products c[k] = a0*...*ak (carry coefficients)
    float c[16];
    c[0] = a[0];
#pragma unroll
    for (int k = 1; k < 16; ++k) c[k] = c[k - 1] * a[k];

    // This lane's row of T: Trow[k] = prod_{u=k+1..row} a[u] (k<=row) else 0.
    float Trow[16];
    float suf = 1.0f;
#pragma unroll
    for (int k = 15; k >= 0; --k) {
      Trow[k] = (k <= row) ? suf : 0.0f;
      suf *= (k <= row) ? a[k] : 1.0f;
    }

    v8f Cacc = {};
#pragma unroll
    for (int kb = 0; kb < 4; ++kb) {
      // A (16x4 f32): lane holds (M=row, K=2*hi+v) in VGPR v
      v2f Av;
      Av.x = hi ? Trow[4 * kb + 2] : Trow[4 * kb + 0];
      Av.y = hi ? Trow[4 * kb + 3] : Trow[4 * kb + 1];
      // B (4x16 f32): lane holds (K=2*hi+v, N=row) in VGPR v
      const int kq0 = 4 * kb + 2 * hi;
      const int kq1 = kq0 + 1;
      const float a0 = hi ? a[4 * kb + 2] : a[4 * kb + 0];
      const float a1 = hi ? a[4 * kb + 3] : a[4 * kb + 1];
      v2f Bv;
      Bv.x = (1.0f - a0) * hidden[hBase + (size_t)(m0 + kq0) * D_ + d0 + row];
      Bv.y = (1.0f - a1) * hidden[hBase + (size_t)(m0 + kq1) * D_ + d0 + row];
      Cacc = __builtin_amdgcn_wmma_f32_16x16x4_f32(
          false, Av, false, Bv, (short)0, Cacc, false, false);
    }

    // Prefetch next chunk's 16x16 hidden tile (global_prefetch_b8)
    if (cb + 1 < SEGCHUNKS_)
      __builtin_prefetch(&hidden[hBase + (size_t)(m0 + 16 + row) * D_ + d0], 0,
                         1);

    // h[m] = Cacc[m] + c[m]*carry  (m = r2 + 8*hi within chunk)
    float h[8];
#pragma unroll
    for (int r2 = 0; r2 < 8; ++r2)
      h[r2] = Cacc[r2] + (hi ? c[r2 + 8] : c[r2]) * carry;

    // New carry = h[15][n]: VGPR7 of lanes 16..31 (n = lane-16)
    float newcarry = __shfl(h[7], 16 + row, 32);

#pragma unroll
    for (int r2 = 0; r2 < 8; ++r2) {
      const int m = m0 + r2 + 8 * hi;
      emaBuf[hBase + (size_t)m * D_ + d0 + row] = h[r2];
    }
    carry = newcarry;
  }

  // segment-final local h per column (lanes 0..15 carry unique columns)
  if (hi == 0)
    segfinal[((size_t)(b * S_ + seg)) * D_ + d0 + row] = carry;
}

// ---------------------------------------------------------------------------
// Phase B2: per (b,d) combine segment finals into per-segment carry-in.
//   carryIn[s] = h[segment s start - 1];  carryIn[0] = state
//   carry_{s+1} = Pseg[s]*carry_s + segfinal[s]
// ---------------------------------------------------------------------------
__global__ void phaseB2_combine(const float* __restrict__ state,
                                const float* __restrict__ Pseg,
                                const float* __restrict__ segfinal,
                                float* __restrict__ carryIn) {
  const int idx = blockIdx.x * blockDim.x + threadIdx.x;  // B_*D_
  const int b = idx / D_;
  const int d = idx % D_;
  float acc = state[idx];
#pragma unroll
  for (int s = 0; s < S_; ++s) {
    const size_t o = ((size_t)(b * S_ + s)) * D_ + d;
    carryIn[o] = acc;
    acc = Pseg[b * S_ + s] * acc + segfinal[o];
  }
}

// ---------------------------------------------------------------------------
// Phase C: out = residual + gathered, carry-fixed chunk state (float4-wide).
//   h(ci) = h_local[b,ci,:] + cumseg[b,ci] * carryIn[b, ci/SEGM, :]
// (gathered ci is always < counts[b], so no validity mask needed)
// ---------------------------------------------------------------------------
__global__ void phaseC_gather_add(const float* __restrict__ residual,
                                  const int* __restrict__ cidxBuf,
                                  const float* __restrict__ emaBuf,
                                  const float* __restrict__ cumseg,
                                  const float* __restrict__ carryIn,
                                  float* __restrict__ out) {
  const size_t q = (size_t)blockIdx.x * blockDim.x + threadIdx.x;  // float4 idx
  const size_t e = q * 4;
  const int d = (int)(e % D_);
  const int l = (int)((e / D_) % L_);
  const int b = (int)(e / ((size_t)D_ * L_));
  float4 o = ((const float4*)residual)[q];
  const int ci = cidxBuf[b * L_ + l];
  if (ci >= 0) {
    const int cc = ci < (M_ - 1) ? ci : (M_ - 1);
    const int s = cc / SEGM_;
    const float cp = cumseg[b * M_ + cc];
    const float4 hl =
        ((const float4*)emaBuf)[(((size_t)b * M_ + cc) * D_ + d) >> 2];
    const float4 cy =
        ((const float4*)carryIn)[(((size_t)(b * S_ + s)) * D_ + d) >> 2];
    o.x += hl.x + cp * cy.x;
    o.y += hl.y + cp * cy.y;
    o.z += hl.z + cp * cy.z;
    o.w += hl.w + cp * cy.w;
  }
  ((float4*)out)[q] = o;
}

// ---------------------------------------------------------------------------
// Phase D: new_state = counts>0 ? fixed-up h at (counts-1) : state
// ---------------------------------------------------------------------------
__global__ void phaseD_state(const float* __restrict__ state,
                             const int* __restrict__ counts,
                             const float* __restrict__ emaBuf,
                             const float* __restrict__ cumseg,
                             const float* __restrict__ carryIn,
                             float* __restrict__ outState) {
  const int idx = blockIdx.x * blockDim.x + threadIdx.x;  // B_*D_ threads
  const int b = idx / D_;
  const int d = idx % D_;
  const int cnt = counts[b];
  float r;
  if (cnt > 0) {
    const int m = cnt - 1;
    const int s = m / SEGM_;
    r = emaBuf[((size_t)b * M_ + m) * D_ + d] +
        cumseg[b * M_ + m] * carryIn[((size_t)(b * S_ + s)) * D_ + d];
  } else {
    r = state[idx];
  }
  outState[idx] = r;
}

extern "C" void kernel_launch(void* const* d_in, const int* in_sizes, int n_in,
                              void* d_out, int out_size, void* d_ws,
                              size_t ws_size, hipStream_t stream) {
  (void)in_sizes; (void)n_in; (void)out_size; (void)ws_size;
  const float* hidden = (const float*)d_in[0];                // (B,M,D) f32
  const float* residual = (const float*)d_in[1];              // (B,L,D) f32
  const float* prob = (const float*)d_in[2];                  // (B,L)   f32
  const float* state = (const float*)d_in[3];                 // (B,D)   f32
  const unsigned char* mask = (const unsigned char*)d_in[4];  // (B,L) bool
  const int* counts = (const int*)d_in[5];                    // (B,)    i32

  float* out = (float*)d_out;                    // (B,L,D) f32
  float* outState = out + (size_t)B_ * L_ * D_;  // (B,D)   f32

  // Workspace layout:
  //   decay   B*M f32 | cumseg B*M f32 | Pseg B*S f32 (padded 4KB)
  // | cidx    B*L i32 | segfinal B*S*D f32 | carryIn B*S*D f32
  // | ema     B*M*D f32
  char* ws = (char*)d_ws;
  size_t off = 0;
  float* decayBuf = (float*)(ws + off);
  off += (size_t)B_ * M_ * sizeof(float);
  float* cumseg = (float*)(ws + off);
  off += (size_t)B_ * M_ * sizeof(float);
  float* Pseg = (float*)(ws + off);
  off += 4096;
  int* cidxBuf = (int*)(ws + off);
  off += (size_t)B_ * L_ * sizeof(int);
  float* segfinal = (float*)(ws + off);
  off += (size_t)B_ * S_ * D_ * sizeof(float);
  float* carryIn = (float*)(ws + off);
  off += (size_t)B_ * S_ * D_ * sizeof(float);
  float* emaBuf = (float*)(ws + off);

  phaseA_compact<<<B_, 256, 0, stream>>>(prob, mask, decayBuf, cidxBuf);
  phaseA2_cumprod<<<1, B_ * S_, 0, stream>>>(decayBuf, cumseg, Pseg);
  phaseB1_ema_wmma<<<B_ * 32 * S_, 32, 0, stream>>>(hidden, decayBuf, emaBuf,
                                                    segfinal);
  phaseB2_combine<<<(B_ * D_) / 256, 256, 0, stream>>>(state, Pseg, segfinal,
                                                       carryIn);
  phaseC_gather_add<<<(B_ * L_ * D_ / 4) / 256, 256, 0, stream>>>(
      residual, cidxBuf, emaBuf, cumseg, carryIn, out);
  phaseD_state<<<(B_ * D_) / 256, 256, 0, stream>>>(state, counts, emaBuf,
                                                    cumseg, carryIn, outState);
}